// DeMOLTaAttention_58609123721693
// MI455X (gfx1250) — compile-verified
//
#include <hip/hip_runtime.h>
#include <hip/hip_bf16.h>
#include <math.h>

// Problem constants (from reference)
#define B_   4
#define S_   512
#define D_   512
#define E_   128
#define H_   16
#define DH_  32

typedef float v2f __attribute__((ext_vector_type(2)));
typedef float v8f __attribute__((ext_vector_type(8)));

// Workspace layout (floats):
//   q    : [B,H,S,DH]  = 1048576
//   k    : [B,H,S,DH]  = 1048576
//   v    : [B,H,S,DH]  = 1048576
//   ksum : [B,H,S]     = 32768
//   qsum : [B,H,S]     = 32768
#define WS_QKV_ONE  (B_ * H_ * S_ * DH_)        // 1048576
#define WS_KSUM_OFF (3 * WS_QKV_ONE)            // 3145728
#define WS_QSUM_OFF (WS_KSUM_OFF + B_ * H_ * S_)

__device__ __forceinline__ v8f wmma_f32(v2f a, v2f b, v8f c) {
    return __builtin_amdgcn_wmma_f32_16x16x4_f32(
        /*neg_a=*/false, a, /*neg_b=*/false, b,
        /*c_mod=*/(short)0, c, /*reuse_a=*/false, /*reuse_b=*/false);
}

// -------------------------------------------------------------------------
// Kernel 1: qkv = x @ Wqkv + bqkv  (M=2048, N=1536, K=512) via WMMA f32.
// Each wave computes a 64(M) x 16(N) strip: 4 accum tiles sharing one B frag.
// Output scattered into q/k/v [B,H,S,DH] with bias fused.
// Each 16-col N-tile sits entirely inside one (head, q/k/v) slice since the
// per-head span is 96 = 6*16 columns.
// -------------------------------------------------------------------------
__global__ __launch_bounds__(256) void qkv_gemm_kernel(
    const float* __restrict__ x, const float* __restrict__ Wqkv,
    const float* __restrict__ bqkv, float* __restrict__ qkv_ws) {
    const int lane = threadIdx.x & 31;
    const int wid  = (blockIdx.x * blockDim.x + threadIdx.x) >> 5; // 0..3071
    const int mt = wid / 96;        // 32 M strips of 64 rows
    const int nt = wid % 96;        // 96 N tiles of 16 cols
    const int r0 = mt * 64;
    const int n0 = nt * 16;
    const int rl = lane & 15;
    const int kb = (lane >> 4) * 2; // K sub-offset per A/B layout

    v8f c0 = {}, c1 = {}, c2 = {}, c3 = {};
    for (int k0 = 0; k0 < D_; k0 += 4) {
        v2f bf;
        bf.x = Wqkv[(k0 + kb) * (3 * D_) + n0 + rl];
        bf.y = Wqkv[(k0 + kb + 1) * (3 * D_) + n0 + rl];
        const float* xp = x + (size_t)(r0 + rl) * D_ + k0 + kb;
        v2f a0 = *(const v2f*)(xp);
        v2f a1 = *(const v2f*)(xp + 16 * D_);
        v2f a2 = *(const v2f*)(xp + 32 * D_);
        v2f a3 = *(const v2f*)(xp + 48 * D_);
        c0 = wmma_f32(a0, bf, c0);
        c1 = wmma_f32(a1, bf, c1);
        c2 = wmma_f32(a2, bf, c2);
        c3 = wmma_f32(a3, bf, c3);
    }

    // Write-out: col -> (head, which of q/k/v, d)
    const int col   = n0 + rl;
    const int h     = col / 96;
    const int inner = col % 96;
    const int which = inner >> 5;   // 0=q, 1=k, 2=v (tile never straddles)
    const int d     = inner & 31;
    const float bias = bqkv[col];
    float* base = qkv_ws + which * WS_QKV_ONE;
    const int rup = (lane >> 4) * 8;

    v8f cc[4] = {c0, c1, c2, c3};
#pragma unroll
    for (int t = 0; t < 4; ++t) {
#pragma unroll
        for (int m = 0; m < 8; ++m) {
            int row = r0 + t * 16 + rup + m;         // global (b*S + s)
            int b = row >> 9, s = row & (S_ - 1);
            base[(((b * H_ + h) * S_) + s) * DH_ + d] = cc[t][m] + bias;
        }
    }
}

// -------------------------------------------------------------------------
// Kernel 2: ksum/qsum over dh for every (b,h,s).
// -------------------------------------------------------------------------
__global__ __launch_bounds__(256) void sums_kernel(float* __restrict__ ws) {
    const int id = blockIdx.x * blockDim.x + threadIdx.x; // 0..32767
    if (id >= B_ * H_ * S_) return;
    const float* qr = ws + (size_t)id * DH_;
    const float* kr = ws + WS_QKV_ONE + (size_t)id * DH_;
    float qa = 0.f, ka = 0.f;
#pragma unroll
    for (int d4 = 0; d4 < DH_ / 4; ++d4) {
        float4 qv = *(const float4*)(qr + d4 * 4);
        float4 kv = *(const float4*)(kr + d4 * 4);
        qa += qv.x + qv.y + qv.z + qv.w;
        ka += kv.x + kv.y + kv.z + kv.w;
    }
    ws[WS_KSUM_OFF + id] = ka;
    ws[WS_QSUM_OFF + id] = qa;
}

// -------------------------------------------------------------------------
// Kernel 3: fused attention, one block per (b, i) query row, 256 threads.
//   scores[h][j] = q·k + dot(p[b,i,j,:], w_h) + bias_h ; mask ; softmax ; @v
// where w_h[e] = Wrqk[e,2h]*ksum[h,i] + Wrqk[e,2h+1]*qsum[h,i].
// p is streamed once with non-temporal loads through WMMA f32 16x16x4.
// -------------------------------------------------------------------------
__global__ __launch_bounds__(256) void attn_kernel(
    const float* __restrict__ p, const int* __restrict__ mask,
    const float* __restrict__ Wrqk, const float* __restrict__ brqk,
    const float* __restrict__ ws, float* __restrict__ out) {
    const int b = blockIdx.x >> 9;
    const int i = blockIdx.x & (S_ - 1);
    const int t = threadIdx.x;
    const int lane = t & 31;
    const int wave = t >> 5;

    __shared__ float qs[H_][DH_];        // q rows for this i, all heads
    __shared__ float wm[E_][H_];         // folded projection weights
    __shared__ float sc[H_ * 513];       // padded score rows (bank-friendly)
    __shared__ float ksh[H_], qsh[H_], bi[H_];

    const float scale = 1.0f / sqrtf(3.0f * (float)DH_);

    // ---- Phase 1a: per-head sums + q rows ----
    if (t < H_) {
        ksh[t] = ws[WS_KSUM_OFF + (b * H_ + t) * S_ + i];
        qsh[t] = ws[WS_QSUM_OFF + (b * H_ + t) * S_ + i];
    }
#pragma unroll
    for (int n = t; n < H_ * DH_; n += 256) {
        int h = n >> 5, d = n & 31;
        qs[h][d] = ws[(((b * H_ + h) * S_) + i) * DH_ + d];
    }
    __syncthreads();

    // ---- Phase 1b: fold weights + bias ----
#pragma unroll
    for (int n = t; n < E_ * H_; n += 256) {
        int e = n >> 4, h = n & 15;
        wm[e][h] = Wrqk[e * (2 * H_) + 2 * h] * ksh[h] +
                   Wrqk[e * (2 * H_) + 2 * h + 1] * qsh[h];
    }
    if (t < H_) bi[t] = brqk[2 * t] * ksh[t] + brqk[2 * t + 1] * qsh[t];
    __syncthreads();

    // ---- Phase 2: q·k (per-head GEMV, k is L2-resident) + bias -> sc ----
    const float* kbase = ws + WS_QKV_ONE;
#pragma unroll 1
    for (int n = t; n < H_ * S_; n += 256) {
        int h = n >> 9, j = n & (S_ - 1);
        const float* kr = kbase + (((b * H_ + h) * S_) + j) * DH_;
        float acc = bi[h];
#pragma unroll
        for (int d4 = 0; d4 < DH_ / 4; ++d4) {
            float4 kv = *(const float4*)(kr + d4 * 4);
            acc += kv.x * qs[h][d4 * 4 + 0];
            acc += kv.y * qs[h][d4 * 4 + 1];
            acc += kv.z * qs[h][d4 * 4 + 2];
            acc += kv.w * qs[h][d4 * 4 + 3];
        }
        sc[h * 513 + j] = acc;
    }
    __syncthreads();

    // ---- Phase 3: p projection via WMMA f32 16x16x4 ----
    // D[16j x 16h] += sum_{e} p[i, j0+jr, e] * wm[e][h]
    const int rl = lane & 15;
    const int kb = (lane >> 4) * 2;
    v2f bf[E_ / 4];
#pragma unroll
    for (int s = 0; s < E_ / 4; ++s) {
        int e = s * 4 + kb;
        bf[s].x = wm[e][rl];
        bf[s].y = wm[e + 1][rl];
    }
    const float* prow = p + (size_t)(b * S_ + i) * S_ * E_;
#pragma unroll 1
    for (int jt = wave; jt < S_ / 16; jt += 8) {
        const int j0 = jt * 16;
        v8f c = {};
        const float* pr = prow + (size_t)(j0 + rl) * E_ + kb;
#pragma unroll
        for (int s = 0; s < E_ / 4; ++s) {
            v2f a = __builtin_nontemporal_load((const v2f*)(pr + s * 4));
            c = wmma_f32(a, bf[s], c);
        }
        // scatter-add into sc: lane -> (h = lane&15, j = j0 + (lane>>4)*8 + m)
        const int hh = lane & 15;
        const int jb = j0 + (lane >> 4) * 8;
#pragma unroll
        for (int m = 0; m < 8; ++m) sc[hh * 513 + jb + m] += c[m];
    }
    __syncthreads();

    // ---- Phase 4: masked softmax per head (2 heads per wave) ----
    const int* mrow = mask + (size_t)(b * S_ + i) * S_;
#pragma unroll
    for (int hi = 0; hi < 2; ++hi) {
        const int h = wave + hi * 8;
        float mx = -3.402823466e+38f;
#pragma unroll
        for (int it = 0; it < S_ / 32; ++it) {
            int j = lane + it * 32;
            float s = (mrow[j] == 0) ? -10000.0f : sc[h * 513 + j];
            s *= scale;
            sc[h * 513 + j] = s;
            mx = fmaxf(mx, s);
        }
#pragma unroll
        for (int off = 16; off >= 1; off >>= 1)
            mx = fmaxf(mx, __shfl_xor(mx, off, 32));
        float sum = 0.f;
#pragma unroll
        for (int it = 0; it < S_ / 32; ++it) {
            int j = lane + it * 32;
            float e = __expf(sc[h * 513 + j] - mx);
            sc[h * 513 + j] = e;
            sum += e;
        }
#pragma unroll
        for (int off = 16; off >= 1; off >>= 1)
            sum += __shfl_xor(sum, off, 32);
        float inv = 1.0f / sum;
#pragma unroll
        for (int it = 0; it < S_ / 32; ++it)
            sc[h * 513 + lane + it * 32] *= inv;
    }
    __syncthreads();

    // ---- Phase 5: out[h][d] = sum_j probs[h][j] * v[b,h,j,d] ----
    const float* vbase = ws + 2 * WS_QKV_ONE;
#pragma unroll
    for (int n = t; n < H_ * DH_; n += 256) {
        int h = n >> 5, d = n & 31;
        const float* vb = vbase + ((size_t)(b * H_ + h) * S_) * DH_ + d;
        const float* pr2 = &sc[h * 513];
        float acc = 0.f;
#pragma unroll 8
        for (int j = 0; j < S_; ++j) acc += pr2[j] * vb[(size_t)j * DH_];
        out[((size_t)(b * S_ + i)) * D_ + h * DH_ + d] = acc;
    }
}

// -------------------------------------------------------------------------
extern "C" void kernel_launch(void* const* d_in, const int* in_sizes, int n_in,
                              void* d_out, int out_size, void* d_ws, size_t ws_size,
                              hipStream_t stream) {
    const float* x    = (const float*)d_in[0];
    const float* p    = (const float*)d_in[1];
    const int*   mask = (const int*)d_in[2];
    const float* Wqkv = (const float*)d_in[3];
    const float* bqkv = (const float*)d_in[4];
    const float* Wrqk = (const float*)d_in[5];
    const float* brqk = (const float*)d_in[6];
    float* out = (float*)d_out;
    float* ws  = (float*)d_ws;

    // Kernel 1: 3072 waves (32 M-strips x 96 N-tiles), 8 waves/block
    qkv_gemm_kernel<<<384, 256, 0, stream>>>(x, Wqkv, bqkv, ws);
    // Kernel 2: 32768 (b,h,s) sums
    sums_kernel<<<(B_ * H_ * S_ + 255) / 256, 256, 0, stream>>>(ws);
    // Kernel 3: one block per (b,i)
    attn_kernel<<<B_ * S_, 256, 0, stream>>>(p, mask, Wrqk, brqk, ws, out);
}